// CapsuleLayer_57475252355332
// MI455X (gfx1250) — compile-verified
//
#include <hip/hip_runtime.h>
#include <cstdint>

#define B_  32
#define I_  64
#define J_  1152
#define K_  32
#define M_  16
#define ROWS (I_*K_)      // 2048 (i,k) rows
#define UHP 33            // padded LDS row stride for 32 cols (conflict-free, gcd(33,64)=1)
#define BLK 512           // 16 waves

typedef float v2f __attribute__((ext_vector_type(2)));
typedef float v8f __attribute__((ext_vector_type(8)));

// One routing pass for one j, full batch (B=32):
//   u_hat tile [2048 x 32] = W_j [2048 x 16] * X_j [16 x 32]   (WMMA f32 16x16x4, into LDS)
//   MODE 0: c = 1/64 (softmax of zeros), accumulate s0
//   MODE 1: agree = <out0, u_hat>_k ; bstate = agree ; c = softmax_i ; accumulate s1
//   MODE 2: agree with out1 ; b2 = bstate + agree ; c = softmax_i ; accumulate s2
template <int MODE>
__global__ __launch_bounds__(BLK)
void caps_pass(const float* __restrict__ xin, const float* __restrict__ W,
               const float* __restrict__ outPrev, float* __restrict__ bstate,
               float* __restrict__ sacc)
{
  __shared__ float uh[ROWS * UHP];  // 270336 B : u_hat tile [row=(i*32+k)][b]
  __shared__ float xs[M_ * B_];     //   2048 B : X tile [m][b]
  __shared__ float bv[I_ * UHP];    //   8448 B : logits -> softmax weights [i][b]

  const int t    = threadIdx.x;
  const int lane = t & 31;
  const int wave = t >> 5;
  const int l15  = lane & 15;
  const int hi   = lane >> 4;       // 0: lanes 0-15, 1: lanes 16-31
  const int j    = blockIdx.x;

  // ---- Phase A: load X[m][b] = inputs[b][j][m] (512 threads == 16m x 32b) ----
  {
    int m = t & 15, bb = t >> 4;
    xs[m * B_ + bb] = xin[((size_t)bb * J_ + j) * M_ + m];
  }
  __syncthreads();

  // ---- Phase B: u_hat tile via WMMA. 128 row-tiles x 2 col-tiles; 16 waves ----
  {
    // B operands are invariant across row-tiles: hoist. bm[ms][ct]
    // 32-bit B 4x16 layout: lanes 0-15 rows {m0,m0+1}, lanes 16-31 rows {m0+2,m0+3}, col = lane%16
    v2f bm0[4], bm1[4];
#pragma unroll
    for (int ms = 0; ms < 4; ++ms) {
      int m0 = ms * 4 + hi * 2;
      bm0[ms].x = xs[(m0 + 0) * B_ + l15];
      bm0[ms].y = xs[(m0 + 1) * B_ + l15];
      bm1[ms].x = xs[(m0 + 0) * B_ + 16 + l15];
      bm1[ms].y = xs[(m0 + 1) * B_ + 16 + l15];
    }
    for (int wt = 0; wt < 8; ++wt) {
      int tr = wave * 8 + wt;             // row-tile 0..127
      int i  = tr >> 1;
      int k  = ((tr & 1) << 4) + l15;     // row = i*32 + k
      const float* wrow = W + (((size_t)i * J_ + j) * K_ + k) * M_;
      v8f acc0 = {0.f,0.f,0.f,0.f,0.f,0.f,0.f,0.f};
      v8f acc1 = {0.f,0.f,0.f,0.f,0.f,0.f,0.f,0.f};
#pragma unroll
      for (int ms = 0; ms < 4; ++ms) {
        int m0 = ms * 4 + hi * 2;         // A lanes 0-15: cols {m0,m0+1}; 16-31: {m0+2,m0+3}
        v2f a = *(const v2f*)(wrow + m0);
        acc0 = __builtin_amdgcn_wmma_f32_16x16x4_f32(false, a, false, bm0[ms],
                                                     (short)0, acc0, false, false);
        acc1 = __builtin_amdgcn_wmma_f32_16x16x4_f32(false, a, false, bm1[ms],
                                                     (short)0, acc1, false, false);
      }
      int rbase = tr * 16 + hi * 8;       // C layout: VGPR r -> M=r (lo lanes) / M=8+r (hi lanes)
#pragma unroll
      for (int r = 0; r < 8; ++r) {
        uh[(rbase + r) * UHP + l15]      = acc0[r];
        uh[(rbase + r) * UHP + 16 + l15] = acc1[r];
      }
    }
  }
  __syncthreads();

  // ---- Phase C: agreement + softmax over i (MODE >= 1) ----
  if (MODE >= 1) {
#pragma unroll
    for (int p = 0; p < 4; ++p) {
      int pair = t + p * BLK;             // 2048 (i,b) pairs
      int bb = pair & 31;
      int i  = pair >> 5;
      const float* op = outPrev + ((size_t)bb * I_ + i) * K_;
      float agree = 0.f;
#pragma unroll
      for (int k = 0; k < K_; ++k) agree += op[k] * uh[(i * K_ + k) * UHP + bb];
      size_t baddr = ((size_t)j * I_ + i) * B_ + bb;   // bstate layout [j][i][b] (coalesced)
      float bval;
      if (MODE == 1) { bval = agree; bstate[baddr] = bval; }
      else           { bval = bstate[baddr] + agree;      }
      bv[i * UHP + bb] = bval;
    }
    __syncthreads();
    if (t < B_) {                          // softmax over i per b column
      int bb = t;
      float mx = -3.402823466e38f;
      for (int i = 0; i < I_; ++i) mx = fmaxf(mx, bv[i * UHP + bb]);
      float sum = 0.f;
      for (int i = 0; i < I_; ++i) {
        float e = __expf(bv[i * UHP + bb] - mx);
        bv[i * UHP + bb] = e;
        sum += e;
      }
      float inv = 1.f / sum;
      for (int i = 0; i < I_; ++i) bv[i * UHP + bb] *= inv;
    }
    __syncthreads();
  }

  // ---- Phase D: s[b,i,k] += c[b,i] * u_hat  (coalesced f32 atomics) ----
  const float cuni = 1.0f / (float)I_;
#pragma unroll
  for (int bb = 0; bb < B_; ++bb) {
    float* srow = sacc + (size_t)bb * ROWS;
#pragma unroll
    for (int q = 0; q < 4; ++q) {
      int r = t + q * BLK;                 // consecutive lanes -> consecutive addresses
      float c = (MODE == 0) ? cuni : bv[(r >> 5) * UHP + bb];
      unsafeAtomicAdd(srow + r, uh[r * UHP + bb] * c);
    }
  }
}

// squash over k (K=32): one wave32 per (b,i)
__global__ __launch_bounds__(32)
void squash_kernel(const float* __restrict__ s, float* __restrict__ out)
{
  size_t g = blockIdx.x;
  int k = threadIdx.x;
  float v = s[g * K_ + k];
  float sq = v * v;
#pragma unroll
  for (int off = 16; off > 0; off >>= 1) sq += __shfl_xor(sq, off, 32);
  float scale = (sq / (1.f + sq)) * rsqrtf(sq + 1e-7f);
  out[g * K_ + k] = v * scale;
}

__global__ __launch_bounds__(256)
void zero_kernel(float* __restrict__ p, int n)
{
  int i = blockIdx.x * 256 + threadIdx.x;
  if (i < n) p[i] = 0.f;
}

extern "C" void kernel_launch(void* const* d_in, const int* in_sizes, int n_in,
                              void* d_out, int out_size, void* d_ws, size_t ws_size,
                              hipStream_t stream)
{
  (void)in_sizes; (void)n_in; (void)out_size; (void)ws_size;
  const float* xin = (const float*)d_in[0];   // inputs [B, J, M]
  const float* W   = (const float*)d_in[1];   // W [I, J, K, M]

  float* wsf = (float*)d_ws;
  const int SN = B_ * I_ * K_;                // 65536
  float* s0  = wsf;
  float* s1  = wsf + SN;
  float* s2  = wsf + 2 * SN;
  float* o0  = wsf + 3 * SN;
  float* o1  = wsf + 4 * SN;
  float* bst = wsf + 5 * SN;                  // [J][I][B] = 2359296 floats

  zero_kernel<<<(3 * SN) / 256, 256, 0, stream>>>(s0, 3 * SN);

  caps_pass<0><<<J_, BLK, 0, stream>>>(xin, W, nullptr, bst, s0);
  squash_kernel<<<B_ * I_, 32, 0, stream>>>(s0, o0);
  caps_pass<1><<<J_, BLK, 0, stream>>>(xin, W, o0, bst, s1);
  squash_kernel<<<B_ * I_, 32, 0, stream>>>(s1, o1);
  caps_pass<2><<<J_, BLK, 0, stream>>>(xin, W, o1, bst, s2);
  squash_kernel<<<B_ * I_, 32, 0, stream>>>(s2, (float*)d_out);
}